// FrequencyQuantizer_61332132987280
// MI455X (gfx1250) — compile-verified
//
#include <hip/hip_runtime.h>

typedef float v2f __attribute__((ext_vector_type(2)));
typedef float v8f __attribute__((ext_vector_type(8)));

#define K_CODES 1024
#define D_DIM   64
#define HW      4096                 // H*W = 64*64
#define N_TOT   131072               // 32*64*64 rows of z_flat
#define Q_ELEMS 8388608              // 32*64*64*64 quantized elements

// ---------------------------------------------------------------------------
// Zero the histogram + loss accumulator in workspace
// ---------------------------------------------------------------------------
__global__ void vq_zero(int* __restrict__ counts, float* __restrict__ loss_acc) {
    int i = blockIdx.x * blockDim.x + threadIdx.x;
    if (i < K_CODES) counts[i] = 0;
    if (i == 0) *loss_acc = 0.0f;
}

// ---------------------------------------------------------------------------
// enorm[k] = ||embedding[k]||^2
// ---------------------------------------------------------------------------
__global__ void vq_enorm(const float* __restrict__ emb, float* __restrict__ enorm) {
    int k = blockIdx.x * blockDim.x + threadIdx.x;
    if (k < K_CODES) {
        float s = 0.0f;
        #pragma unroll
        for (int d = 0; d < D_DIM; ++d) {
            float e = emb[k * D_DIM + d];
            s += e * e;
        }
        enorm[k] = s;
    }
}

// ---------------------------------------------------------------------------
// Main kernel: WMMA f32 16x16x4 distance matmul + argmin + gather/transpose
// + loss partial sum + index histogram.
// Block = 256 threads = 8 waves; each wave owns 16 rows; block covers 128 rows.
// ---------------------------------------------------------------------------
__global__ __launch_bounds__(256) void vq_main(const float* __restrict__ z,
                                               const float* __restrict__ emb,
                                               const float* __restrict__ enorm,
                                               float* __restrict__ out,
                                               int* __restrict__ counts,
                                               float* __restrict__ loss_acc) {
    __shared__ int   s_idx[128];
    __shared__ float s_loss;

    const int tid  = threadIdx.x;
    const int lane = tid & 31;
    const int warp = tid >> 5;
    const int m    = lane & 15;      // row-within-tile (A) / column (B, C/D)
    const int half = lane >> 4;      // K-offset group for A/B, +8 row group for C/D

    const int blockRow = blockIdx.x * 128;
    const int waveRow  = blockRow + warp * 16;
    const int n        = waveRow + m;        // this lane's z_flat row
    const int b        = n >> 12;            // n / 4096 (never splits inside a wave tile)
    const int hw       = n & 4095;

    if (tid == 0) s_loss = 0.0f;

    // --- Load A fragments: z_flat[n][d], d = 4*kk + 2*half (+1). z is [B,C,H,W]
    // so z_flat[n][d] = z[b*64*4096 + d*4096 + hw]; lanes 0-15 share d ->
    // 64B-contiguous across lanes. 16 rows x 64 D live in 32 VGPRs, reused later.
    const float* zrow = z + (size_t)b * D_DIM * HW + hw;
    v2f a[16];
    #pragma unroll
    for (int kk = 0; kk < 16; ++kk) {
        const int d0 = 4 * kk + 2 * half;
        a[kk].x = zrow[(size_t)d0 * HW];
        a[kk].y = zrow[(size_t)(d0 + 1) * HW];
    }

    // Running argmin state: acc[v] = score[row = v + 8*half][col = kbase + m]
    float bestd[8];
    int   besti[8];
    #pragma unroll
    for (int v = 0; v < 8; ++v) { bestd[v] = 3.4e38f; besti[v] = 0; }

    // --- Sweep K in 16-column chunks: 16 chained WMMA (K=4 each) cover D=64.
    for (int kbase = 0; kbase < K_CODES; kbase += 16) {
        const int col = kbase + m;
        const float* erow = emb + (size_t)col * D_DIM + 2 * half;

        v2f bf[16];
        #pragma unroll
        for (int kk = 0; kk < 16; ++kk)
            bf[kk] = *(const v2f*)(erow + 4 * kk);   // B: K = 4*kk + 2*half (+1), N = col

        v8f acc = {0.f, 0.f, 0.f, 0.f, 0.f, 0.f, 0.f, 0.f};
        #pragma unroll
        for (int kk = 0; kk < 16; ++kk)
            acc = __builtin_amdgcn_wmma_f32_16x16x4_f32(
                      false, a[kk], false, bf[kk], (short)0, acc, false, false);

        const float en = enorm[col];
        #pragma unroll
        for (int v = 0; v < 8; ++v) {
            const float dist = en - 2.0f * acc[v];
            if (dist < bestd[v]) { bestd[v] = dist; besti[v] = col; }
        }
    }

    // --- Min-reduce across the 16 lanes sharing each row (ties -> lower index)
    #pragma unroll
    for (int off = 1; off < 16; off <<= 1) {
        #pragma unroll
        for (int v = 0; v < 8; ++v) {
            const float od = __shfl_xor(bestd[v], off, 32);
            const int   oi = __shfl_xor(besti[v], off, 32);
            if (od < bestd[v] || (od == bestd[v] && oi < besti[v])) {
                bestd[v] = od; besti[v] = oi;
            }
        }
    }
    // lanes 0..7 hold rows 0..7, lanes 16..23 hold rows 8..15
    if (m < 8) s_idx[warp * 16 + half * 8 + m] = besti[m];
    __syncthreads();

    // --- Indices output (as float) + histogram
    if (tid < 128) {
        const int idx = s_idx[tid];
        out[(size_t)Q_ELEMS + 2 + blockRow + tid] = (float)idx;
        atomicAdd(&counts[idx], 1);
    }

    // --- Gather winning code row, write transposed quantized output, and
    //     accumulate (q - z)^2 reusing the resident A fragments.
    const int idx = s_idx[warp * 16 + m];
    const float* qe = emb + (size_t)idx * D_DIM;
    float* outrow = out + (size_t)b * D_DIM * HW + hw;
    float sq = 0.0f;
    #pragma unroll
    for (int kk = 0; kk < 16; ++kk) {
        const int d0 = 4 * kk + 2 * half;
        const float ex = qe[d0];
        const float ey = qe[d0 + 1];
        outrow[(size_t)d0 * HW]       = ex;
        outrow[(size_t)(d0 + 1) * HW] = ey;
        const float dx = ex - a[kk].x;
        const float dy = ey - a[kk].y;
        sq += dx * dx + dy * dy;
    }
    atomicAdd(&s_loss, sq);
    __syncthreads();
    if (tid == 0) atomicAdd(loss_acc, s_loss);
}

// ---------------------------------------------------------------------------
// Finalize: loss scalar + perplexity from histogram
// ---------------------------------------------------------------------------
__global__ __launch_bounds__(1024) void vq_final(const int* __restrict__ counts,
                                                 const float* __restrict__ loss_acc,
                                                 float* __restrict__ out) {
    __shared__ float red[1024];
    const int k = threadIdx.x;
    const float p = (float)counts[k] / (float)N_TOT;
    red[k] = p * logf(p + 1e-10f);
    __syncthreads();
    for (int s = 512; s > 0; s >>= 1) {
        if (k < s) red[k] += red[k + s];
        __syncthreads();
    }
    if (k == 0) {
        out[Q_ELEMS]     = 1.25f * (*loss_acc) / (float)(N_TOT * D_DIM);  // q + 0.25*e loss
        out[Q_ELEMS + 1] = expf(-red[0]);                                 // perplexity
    }
}

// ---------------------------------------------------------------------------
extern "C" void kernel_launch(void* const* d_in, const int* in_sizes, int n_in,
                              void* d_out, int out_size, void* d_ws, size_t ws_size,
                              hipStream_t stream) {
    const float* z   = (const float*)d_in[0];   // [32, 64, 64, 64] f32
    const float* emb = (const float*)d_in[1];   // [1024, 64] f32
    float* out = (float*)d_out;

    float* enorm    = (float*)d_ws;                       // 1024 f32
    int*   counts   = (int*)((char*)d_ws + 4096);         // 1024 i32
    float* loss_acc = (float*)((char*)d_ws + 8192);       // 1 f32

    vq_zero <<<1, 1024, 0, stream>>>(counts, loss_acc);
    vq_enorm<<<4,  256, 0, stream>>>(emb, enorm);
    vq_main <<<N_TOT / 128, 256, 0, stream>>>(z, emb, enorm, out, counts, loss_acc);
    vq_final<<<1, 1024, 0, stream>>>(counts, loss_acc, out);
}